// HeteroConv_26104811225143
// MI455X (gfx1250) — compile-verified
//
#include <hip/hip_runtime.h>

typedef __attribute__((ext_vector_type(2))) float v2f;
typedef __attribute__((ext_vector_type(8))) float v8f;

#define FDIM 128
#define NTYPE 4

// ---------------------------------------------------------------------------
// Zero the agg + cnt region of the workspace (must run every launch: harness
// poisons ws once and never re-poisons, and scatter accumulates atomically).
// Region is contiguous: [agg: 4*N*128 floats][cnt: 4*N floats] = N*516 floats
// = N*129 float4s exactly.
// ---------------------------------------------------------------------------
__global__ void zero_ws_kernel(float4* __restrict__ p, long long n4) {
  long long i = (long long)blockIdx.x * blockDim.x + threadIdx.x;
  if (i < n4) p[i] = make_float4(0.f, 0.f, 0.f, 0.f);
}

// ---------------------------------------------------------------------------
// Build the fused B matrix (640x128) and averaged bias (128):
//   rows   0..127 : mean_t W_self[t]
//   rows 128..639 : 0.25 * W_neigh[t]   (t = (row-128)/128)
// ---------------------------------------------------------------------------
__global__ void prep_weights_kernel(const float* __restrict__ Wself,
                                    const float* __restrict__ Wneigh,
                                    const float* __restrict__ b,
                                    float* __restrict__ Wc,
                                    float* __restrict__ bavg) {
  int idx = blockIdx.x * blockDim.x + threadIdx.x;
  if (idx >= 640 * FDIM) return;
  int row = idx >> 7;
  int col = idx & 127;
  float v;
  if (row < FDIM) {
    v = 0.25f * (Wself[0 * FDIM * FDIM + row * FDIM + col] +
                 Wself[1 * FDIM * FDIM + row * FDIM + col] +
                 Wself[2 * FDIM * FDIM + row * FDIM + col] +
                 Wself[3 * FDIM * FDIM + row * FDIM + col]);
  } else {
    int t  = (row - FDIM) >> 7;
    int rr = (row - FDIM) & 127;
    v = 0.25f * Wneigh[t * FDIM * FDIM + rr * FDIM + col];
  }
  Wc[idx] = v;
  if (idx < FDIM) {
    bavg[idx] = 0.25f * (b[0 * FDIM + idx] + b[1 * FDIM + idx] +
                         b[2 * FDIM + idx] + b[3 * FDIM + idx]);
  }
}

// ---------------------------------------------------------------------------
// Per-edge scatter: one wave32 per edge. Each lane gathers a float4 of the
// source row (x is L2-resident: 25.6MB << 192MB L2) and issues 4 native fp32
// atomic adds into agg[type][dst]. Lane 0 bumps the degree counter.
// ---------------------------------------------------------------------------
__device__ __forceinline__ void atomic_add_f32(float* p, float v) {
  __hip_atomic_fetch_add(p, v, __ATOMIC_RELAXED, __HIP_MEMORY_SCOPE_AGENT);
}

__global__ void scatter_kernel(const float* __restrict__ x,
                               const int* __restrict__ edge_index, // [2*E]
                               const int* __restrict__ edge_type,  // [E]
                               float* __restrict__ agg,            // [4,N,128]
                               float* __restrict__ cnt,            // [4,N]
                               int E, int N) {
  long long gtid = (long long)blockIdx.x * blockDim.x + threadIdx.x;
  int e    = (int)(gtid >> 5);
  int lane = threadIdx.x & 31;
  if (e >= E) return;
  int src = edge_index[e];
  int dst = edge_index[(long long)E + e];
  int t   = edge_type[e];
  const float4 v = ((const float4*)(x + (long long)src * FDIM))[lane];
  float* dp = agg + ((long long)t * N + dst) * FDIM + lane * 4;
  atomic_add_f32(dp + 0, v.x);
  atomic_add_f32(dp + 1, v.y);
  atomic_add_f32(dp + 2, v.z);
  atomic_add_f32(dp + 3, v.w);
  if (lane == 0) atomic_add_f32(cnt + (long long)t * N + dst, 1.0f);
}

// ---------------------------------------------------------------------------
// mean_agg = agg / max(cnt, 1)   (in place)
// ---------------------------------------------------------------------------
__global__ void normalize_kernel(float* __restrict__ agg,
                                 const float* __restrict__ cnt,
                                 long long total) {
  long long i = (long long)blockIdx.x * blockDim.x + threadIdx.x;
  if (i >= total) return;
  float c = cnt[i >> 7];
  agg[i] = agg[i] / fmaxf(c, 1.0f);
}

// ---------------------------------------------------------------------------
// Fused GEMM:  out[N,128] = [x | ma0 | ma1 | ma2 | ma3] (N x 640)
//                          @ Wc (640 x 128)  + bavg
// One wave per 16x16 output tile; V_WMMA_F32_16X16X4_F32, K-loop of 32 per
// 128-wide segment, 5 segments (160 WMMA per tile).
//
// f32 16x4 A fragment layout (ISA 7.12.2): lane L holds row M=L&15;
// VGPR0/1 = K = {2*kh, 2*kh+1} with kh = L>>4.  B (4x16) mirrors this with
// lane = column N.  C/D: VGPR r -> row r + 8*kh, lane -> column.
// ---------------------------------------------------------------------------
__global__ void
__launch_bounds__(256)
hetero_gemm_kernel(const float* __restrict__ x,
                   const float* __restrict__ agg,  // normalized, [4,N,128]
                   const float* __restrict__ Wc,   // [640,128]
                   const float* __restrict__ bavg, // [128]
                   float* __restrict__ out,        // [N,128]
                   int N) {
  const int lane    = threadIdx.x & 31;
  const int tile    = blockIdx.x * 8 + (threadIdx.x >> 5); // 8 waves / block
  const int rowTile = tile >> 3;   // 128/16 = 8 column tiles
  const int colTile = tile & 7;
  const int row0    = rowTile * 16;
  if (row0 >= N) return;           // wave-uniform: EXEC stays all-ones for WMMA

  const int m    = lane & 15;      // row of A / column of B,D
  const int kh   = lane >> 4;      // K-half selector
  const int col0 = colTile * 16;
  const int arow = min(row0 + m, N - 1);

  v8f acc;
  const float bv = bavg[col0 + m];
#pragma unroll
  for (int i = 0; i < 8; ++i) acc[i] = bv;

#pragma unroll 1
  for (int s = 0; s < 5; ++s) {
    const float* Abase = (s == 0) ? x : (agg + (long long)(s - 1) * N * FDIM);
    const float* A = Abase + (long long)arow * FDIM + 2 * kh;
    const float* B = Wc + (long long)s * FDIM * FDIM + (long long)(2 * kh) * FDIM
                        + col0 + m;
#pragma unroll
    for (int k = 0; k < FDIM; k += 4) {
      v2f a = *(const v2f*)(A + k);              // K = k+2kh, k+2kh+1 (8B load)
      v2f bf;
      bf.x = B[(long long)k * FDIM];             // row k+2kh   of B
      bf.y = B[(long long)(k + 1) * FDIM];       // row k+2kh+1 of B
      acc = __builtin_amdgcn_wmma_f32_16x16x4_f32(
          /*neg_a=*/false, a, /*neg_b=*/false, bf,
          /*c_mod=*/(short)0, acc, /*reuse_a=*/false, /*reuse_b=*/false);
    }
  }

  float* O = out + (long long)row0 * FDIM + col0;
#pragma unroll
  for (int r = 0; r < 8; ++r) {
    int rr = r + 8 * kh;
    if (row0 + rr < N) O[(long long)rr * FDIM + m] = acc[r];
  }
}

// ---------------------------------------------------------------------------
extern "C" void kernel_launch(void* const* d_in, const int* in_sizes, int n_in,
                              void* d_out, int out_size, void* d_ws, size_t ws_size,
                              hipStream_t stream) {
  const float* x      = (const float*)d_in[0];
  const int*   ei     = (const int*)d_in[1];   // [2,E]
  const int*   et     = (const int*)d_in[2];   // [E]
  const float* Wself  = (const float*)d_in[3]; // [4,128,128]
  const float* Wneigh = (const float*)d_in[4]; // [4,128,128]
  const float* bias   = (const float*)d_in[5]; // [4,128]
  float*       out    = (float*)d_out;

  const int N = in_sizes[0] / FDIM;
  const int E = in_sizes[2];

  // Workspace layout (floats):
  //   agg  : 4*N*128
  //   cnt  : 4*N            (contiguous with agg for the zero pass)
  //   Wc   : 640*128
  //   bavg : 128
  float* ws   = (float*)d_ws;
  float* agg  = ws;
  float* cnt  = agg + (long long)NTYPE * N * FDIM;
  float* Wc   = cnt + (long long)NTYPE * N;
  float* bavg = Wc + 640 * FDIM;

  // 1) zero agg+cnt : N*516 floats = N*129 float4 exactly
  {
    long long n4 = (long long)N * 129;
    int blocks = (int)((n4 + 255) / 256);
    zero_ws_kernel<<<blocks, 256, 0, stream>>>((float4*)agg, n4);
  }

  // 2) fold weights: Wc (640x128) and bavg (128)
  {
    int total = 640 * FDIM;
    prep_weights_kernel<<<(total + 255) / 256, 256, 0, stream>>>(
        Wself, Wneigh, bias, Wc, bavg);
  }

  // 3) scatter: one wave per edge, 8 edges per 256-thread block
  {
    int blocks = (E + 7) / 8;
    scatter_kernel<<<blocks, 256, 0, stream>>>(x, ei, et, agg, cnt, E, N);
  }

  // 4) normalize agg by counts
  {
    long long total = (long long)NTYPE * N * FDIM;
    int blocks = (int)((total + 255) / 256);
    normalize_kernel<<<blocks, 256, 0, stream>>>(agg, cnt, total);
  }

  // 5) fused WMMA GEMM + bias
  {
    int rowTiles = (N + 15) / 16;
    int tiles    = rowTiles * 8;         // 8 col tiles (F=128)
    int blocks   = (tiles + 7) / 8;      // 8 waves per block
    hetero_gemm_kernel<<<blocks, 256, 0, stream>>>(x, agg, Wc, bavg, out, N);
  }
}